// AR_model_37331855736869
// MI455X (gfx1250) — compile-verified
//
#include <hip/hip_runtime.h>
#include <hip/hip_bf16.h>

typedef float v2f __attribute__((ext_vector_type(2)));
typedef float v8f __attribute__((ext_vector_type(8)));

#define T_STEPS     2048
#define ROWS_PER_WG 16
#define HSTRIDE     4100          /* 2*T + 4 pad: 16B-aligned rows, conflict-free bank skew */
#define KBLK        16            /* outputs per block step (WMMA N) */
#define NWAVES      8
#define WG_THREADS  (NWAVES * 32)
#define KSPAN       (T_STEPS / NWAVES)   /* 256 k-values per wave */
#define NMMA        (KSPAN / 4)          /* 64 WMMAs per wave per block */
#define P_GUARD     16            /* leading zeros implement the j<r band of W */
#define P_SIZE      (P_GUARD + T_STEPS)

#define LDS_HIST_F  (ROWS_PER_WG * HSTRIDE)
#define LDS_P_OFF   LDS_HIST_F
#define LDS_SCR_OFF (LDS_P_OFF + P_SIZE)
#define LDS_TOTAL_F (LDS_SCR_OFF + NWAVES * 256)

__global__ __launch_bounds__(WG_THREADS) void ar_wmma_f32_kernel(
    const float* __restrict__ inputs,
    const float* __restrict__ params,
    const float* __restrict__ bias,
    float* __restrict__ out)
{
    extern __shared__ float smem[];
    float* hist = smem;                 // [16][4100]; [0,2048) seeds, [2048,4096) generated
    float* pbuf = smem + LDS_P_OFF;     // [16 zeros | p_rev[0..2047]]
    float* scr  = smem + LDS_SCR_OFF;   // [NWAVES][32 lanes][8] partial C tiles

    const int tid     = threadIdx.x;
    const int lane    = tid & 31;
    const int wave    = tid >> 5;
    const int rowBase = blockIdx.x * ROWS_PER_WG;

    // ---- stage reversed params with guard zeros ----
    if (tid < P_GUARD) pbuf[tid] = 0.0f;
    for (int m = tid; m < T_STEPS; m += WG_THREADS)
        pbuf[P_GUARD + m] = params[T_STEPS - 1 - m];  // p_rev[m]

    // ---- stage the 16 seed rows (float4, coalesced) ----
    const float4* in4 = reinterpret_cast<const float4*>(inputs);
    for (int idx = tid; idx < ROWS_PER_WG * (T_STEPS / 4); idx += WG_THREADS) {
        int r  = idx >> 9;          // 512 float4-chunks per row
        int c4 = idx & 511;
        float4 v = in4[(size_t)(rowBase + r) * (T_STEPS / 4) + c4];
        *reinterpret_cast<float4*>(&hist[r * HSTRIDE + c4 * 4]) = v;
    }
    __syncthreads();

    const float bval = bias[0];

    // per-lane constant indexing for WMMA A/B tiles
    const int mrow = lane & 15;            // A row / B column
    const int kk   = (lane >> 4) << 1;     // K sub-offset {0,2}
    const int waveKo = wave * KSPAN;
    const float* histRow = &hist[mrow * HSTRIDE + kk];
    const float* pBW     = &pbuf[P_GUARD + kk - mrow + waveKo]; // pBW[j] = W[waveKo+j+kk][mrow]

    // filter taps for the triangular in-block fix-up (params[0..14])
    float pk[KBLK - 1];
#pragma unroll
    for (int k = 0; k < KBLK - 1; ++k)
        pk[k] = pbuf[P_GUARD + T_STEPS - 1 - k];

    int blkOff = 0;  // window start (= blk * 16)

#pragma unroll 1
    for (int blk = 0; blk < T_STEPS / KBLK; ++blk) {
        const float* aPtr = histRow + blkOff + waveKo;

        // ---- 64 x V_WMMA_F32_16X16X4_F32, 4 accumulators, 4-deep operand pipeline ----
        v8f acc[4] = {};
        v2f aReg[4], bReg[4];
#pragma unroll
        for (int s = 0; s < 4; ++s) {
            aReg[s] = *reinterpret_cast<const v2f*>(aPtr + 4 * s);
            bReg[s].x = pBW[4 * s];
            bReg[s].y = pBW[4 * s + 1];
        }
#pragma unroll
        for (int k = 0; k < NMMA; ++k) {
            v2f a = aReg[k & 3];
            v2f b = bReg[k & 3];
            if (k + 4 < NMMA) {
                aReg[k & 3] = *reinterpret_cast<const v2f*>(aPtr + 4 * (k + 4));
                bReg[k & 3].x = pBW[4 * (k + 4)];
                bReg[k & 3].y = pBW[4 * (k + 4) + 1];
            }
            acc[k & 3] = __builtin_amdgcn_wmma_f32_16x16x4_f32(
                false, a, false, b, (short)0, acc[k & 3], false, false);
        }
        v8f cs = (acc[0] + acc[1]) + (acc[2] + acc[3]);

        // ---- spill partial tile as 2 x ds_store_b128: scr[wave][lane][0..7] ----
        float* myScr = &scr[(wave << 8) + (lane << 3)];
        *reinterpret_cast<float4*>(myScr)     = make_float4(cs[0], cs[1], cs[2], cs[3]);
        *reinterpret_cast<float4*>(myScr + 4) = make_float4(cs[4], cs[5], cs[6], cs[7]);
        __syncthreads();

        // ---- wave 0, lane m (<16): reduce 8 partials + triangular fix-up for row m ----
        if (wave == 0 && lane < 16) {
            const int vIdx   = lane & 7;          // C VGPR index of row m
            const int cBase  = (lane >> 3) << 4;  // lane-half offset for row m
            float xs[KBLK];
#pragma unroll
            for (int r = 0; r < KBLK; ++r) {
                const int cOff = ((cBase + r) << 3) + vIdx;
                float acc2 = bval;
#pragma unroll
                for (int w = 0; w < NWAVES; ++w)
                    acc2 += scr[(w << 8) + cOff];
#pragma unroll
                for (int k = 0; k < r; ++k)
                    acc2 += pk[k] * xs[r - 1 - k];
                xs[r] = acc2;
            }
            // append the 16 new values to the linear history; emit to global (float4)
            float4* hrow4 = reinterpret_cast<float4*>(&hist[lane * HSTRIDE + T_STEPS + blkOff]);
            float4* orow4 = reinterpret_cast<float4*>(out + (size_t)(rowBase + lane) * T_STEPS + blkOff);
#pragma unroll
            for (int q = 0; q < 4; ++q) {
                float4 v = make_float4(xs[4 * q], xs[4 * q + 1], xs[4 * q + 2], xs[4 * q + 3]);
                hrow4[q] = v;
                orow4[q] = v;
            }
        }
        __syncthreads();

        blkOff += KBLK;
    }
}

extern "C" void kernel_launch(void* const* d_in, const int* in_sizes, int n_in,
                              void* d_out, int out_size, void* d_ws, size_t ws_size,
                              hipStream_t stream) {
    const float* inputs = (const float*)d_in[0];
    const float* params = (const float*)d_in[1];
    const float* bias   = (const float*)d_in[2];
    float* out = (float*)d_out;

    const int Brows = in_sizes[0] / T_STEPS;                   // 4096
    const size_t shmem = (size_t)LDS_TOTAL_F * sizeof(float);  // ~272 KB < 320 KB/WGP

    (void)hipFuncSetAttribute((const void*)ar_wmma_f32_kernel,
                              hipFuncAttributeMaxDynamicSharedMemorySize,
                              (int)shmem);

    dim3 grid(Brows / ROWS_PER_WG);
    dim3 block(WG_THREADS);
    hipLaunchKernelGGL(ar_wmma_f32_kernel, grid, block, shmem, stream,
                       inputs, params, bias, out);
}